// RAG4DyG_9113920602309
// MI455X (gfx1250) — compile-verified
//
#include <hip/hip_runtime.h>
#include <hip/hip_bf16.h>
#include <math.h>

// ---------------- problem constants (from reference) ----------------
#define BSZ   4096
#define NBR   50
#define NFEAT 128
#define DDIM  64
#define TDIM  16
#define M1    (BSZ * NBR)       // 204800 flattened (b, nb) rows
#define WPAD  66                // padded K stride for LDS tiles (even -> 8B aligned v2f)

typedef float v2f __attribute__((ext_vector_type(2)));
typedef float v8f __attribute__((ext_vector_type(8)));

static __device__ __forceinline__ v8f v8zero() {
    v8f z = {0.f, 0.f, 0.f, 0.f, 0.f, 0.f, 0.f, 0.f};
    return z;
}

// D = A(16x4,f32) * B(4x16,f32) + C(16x16,f32), wave32 WMMA
static __device__ __forceinline__ v8f wmma4(v2f a, v2f b, v8f c) {
    return __builtin_amdgcn_wmma_f32_16x16x4_f32(
        /*neg_a=*/false, a, /*neg_b=*/false, b,
        /*c_mod=*/(short)0, c, /*reuse_a=*/false, /*reuse_b=*/false);
}

static __device__ __forceinline__ float sigmoidf_(float x) {
    return 1.0f / (1.0f + __expf(-x));
}

// =====================================================================
// Kernel A: gather + (node/edge/time linear) + co/sim rank-1 + biases
// One wave handles one 16-row tile of the flattened [M1, *] problem.
// fused[row, d] written to workspace. blockDim = 128 (4 waves).
// =====================================================================
__global__ void __launch_bounds__(128)
embed_fuse_kernel(const float* __restrict__ node_raw,
                  const float* __restrict__ edge_raw,
                  const int*   __restrict__ nids,
                  const int*   __restrict__ eids,
                  const float* __restrict__ times,
                  const int*   __restrict__ dst_ids,
                  const float* __restrict__ Wf, const float* __restrict__ bf,
                  const float* __restrict__ We, const float* __restrict__ be,
                  const float* __restrict__ Wt, const float* __restrict__ bt,
                  const float* __restrict__ Ws, const float* __restrict__ bs,
                  const float* __restrict__ tw, const float* __restrict__ tb,
                  float* __restrict__ fused)
{
    const int lane = threadIdx.x & 31;
    const int wv   = threadIdx.x >> 5;
    const int l15  = lane & 15;
    const int half = lane >> 4;          // 0: K 0..1, 1: K 2..3 of each k-step
    const int koff = 2 * half;

    const int tile    = blockIdx.x * 4 + wv;
    const int rowbase = tile * 16;

    // A-operand role: lane l15 owns matrix row M = l15 (both halves)
    const int arow = rowbase + l15;
    const long long nid = nids[arow];
    const long long eid = eids[arow];
    const float* xn = node_raw + nid * NFEAT;
    const float* xe = edge_raw + eid * NFEAT;

    v8f acc[4];
#pragma unroll
    for (int nt = 0; nt < 4; ++nt) acc[nt] = v8zero();

    // ---- node_emb: X[16x128] @ Wf^T[128x64] ----
    for (int kb = 0; kb < NFEAT; kb += 4) {
        v2f a = *(const v2f*)(xn + kb + koff);
#pragma unroll
        for (int nt = 0; nt < 4; ++nt) {
            v2f b = *(const v2f*)(Wf + (nt * 16 + l15) * NFEAT + kb + koff);
            acc[nt] = wmma4(a, b, acc[nt]);
        }
    }
    // ---- edge_emb accumulated into same tiles ----
    for (int kb = 0; kb < NFEAT; kb += 4) {
        v2f a = *(const v2f*)(xe + kb + koff);
#pragma unroll
        for (int nt = 0; nt < 4; ++nt) {
            v2f b = *(const v2f*)(We + (nt * 16 + l15) * NFEAT + kb + koff);
            acc[nt] = wmma4(a, b, acc[nt]);
        }
    }
    // ---- time_emb: cos(t*w + b)[16x16] @ Wt^T[16x64] ----
    {
        const float t = times[arow];
        for (int kb = 0; kb < TDIM; kb += 4) {
            const int k = kb + koff;
            v2f a;
            a.x = cosf(t * tw[k]     + tb[k]);
            a.y = cosf(t * tw[k + 1] + tb[k + 1]);
#pragma unroll
            for (int nt = 0; nt < 4; ++nt) {
                v2f b = *(const v2f*)(Wt + (nt * 16 + l15) * TDIM + k);
                acc[nt] = wmma4(a, b, acc[nt]);
            }
        }
    }

    // ---- co/sim flags for the 8 rows this lane owns in the D layout ----
    float sflag[8];
#pragma unroll
    for (int r = 0; r < 8; ++r) {
        const int m   = r + 8 * half;         // D layout: VGPR r, half -> M
        const int row = rowbase + m;
        const int nr  = nids[row];
        const int b   = row / NBR;
        const int did = dst_ids[b];
        const float co  = (nr == did) ? 1.0f : 0.0f;
        const int sk = (int)node_raw[(long long)nr  * NFEAT];
        const int qs = (int)node_raw[(long long)did * NFEAT];
        const float sim = (sk == qs) ? 1.0f : 0.0f;
        sflag[r] = co + sim;
    }

    // ---- biases + rank-1 struct terms, store fused tile ----
#pragma unroll
    for (int nt = 0; nt < 4; ++nt) {
        const int d = nt * 16 + l15;          // D layout: lane -> N
        const float bias = bf[d] + be[d] + bt[d] + 2.0f * bs[d];
        const float wsv  = Ws[d];
#pragma unroll
        for (int r = 0; r < 8; ++r) {
            const int row = rowbase + r + 8 * half;
            fused[(long long)row * DDIM + d] = acc[nt][r] + bias + sflag[r] * wsv;
        }
    }
}

// =====================================================================
// Kernel B: sequential GRU over 50 steps + output projection (src).
// One wave owns 16 batches. Wih/Whh staged once in LDS (padded).
// h bounces through per-wave LDS tile for D-layout -> A-layout.
// blockDim = 128 (4 waves), dynamic LDS.
// =====================================================================
#define WLDS_FLOATS (384 * WPAD)   // Wih (192 rows) + Whh (192 rows), K padded

__global__ void __launch_bounds__(128)
gru_out_kernel(const float* __restrict__ fused,
               const float* __restrict__ Wih, const float* __restrict__ bih,
               const float* __restrict__ Whh, const float* __restrict__ bhh,
               const float* __restrict__ Wo,  const float* __restrict__ bo,
               float* __restrict__ out_src)
{
    extern __shared__ float lds[];
    const int tid  = threadIdx.x;
    const int lane = tid & 31;
    const int wv   = tid >> 5;
    const int l15  = lane & 15;
    const int half = lane >> 4;
    const int koff = 2 * half;

    // stage GRU weights into LDS: lds[g*WPAD + k], g in [0,384)
    for (int i = tid; i < 384 * DDIM; i += blockDim.x) {
        const int g = i >> 6, k = i & 63;
        lds[g * WPAD + k] = (g < 192) ? Wih[g * DDIM + k]
                                      : Whh[(g - 192) * DDIM + k];
    }
    float* hld = lds + WLDS_FLOATS + wv * (16 * WPAD);
    for (int i = lane; i < 16 * WPAD; i += 32) hld[i] = 0.0f;  // h0 = 0
    __syncthreads();

    const int base = (blockIdx.x * 4 + wv) * 16;   // batch tile base

    // per-lane gate biases (index d = nt*16 + l15)
    float brz[8], bin_[4], bhn_[4];
#pragma unroll
    for (int nt = 0; nt < 4; ++nt) {
        const int d = nt * 16 + l15;
        brz[nt]     = bih[d]          + bhh[d];
        brz[4 + nt] = bih[DDIM + d]   + bhh[DDIM + d];
        bin_[nt]    = bih[2 * DDIM + d];
        bhn_[nt]    = bhh[2 * DDIM + d];
    }

    v8f h[4];
#pragma unroll
    for (int nt = 0; nt < 4; ++nt) h[nt] = v8zero();

    for (int t = 0; t < NBR; ++t) {
        v8f arz[8], gxn[4], ghn[4];
#pragma unroll
        for (int i = 0; i < 8; ++i) arz[i] = v8zero();
#pragma unroll
        for (int i = 0; i < 4; ++i) { gxn[i] = v8zero(); ghn[i] = v8zero(); }

        const float* fb = fused + ((long long)(base + l15) * NBR + t) * DDIM;

        for (int kb = 0; kb < DDIM; kb += 4) {
            const int k = kb + koff;
            v2f af = *(const v2f*)(fb + k);                 // fused x_t (global)
            v2f ah = *(const v2f*)(hld + l15 * WPAD + k);   // h_{t-1} (LDS)
#pragma unroll
            for (int nt = 0; nt < 12; ++nt) {
                const int g = nt * 16 + l15;
                v2f bi = *(const v2f*)(lds + g * WPAD + k);           // Wih row g
                v2f bh = *(const v2f*)(lds + (192 + g) * WPAD + k);   // Whh row g
                if (nt < 8) {          // r,z gates: gx+gh share accumulator
                    arz[nt] = wmma4(af, bi, arz[nt]);
                    arz[nt] = wmma4(ah, bh, arz[nt]);
                } else {               // n gate: keep gx_n / gh_n separate
                    gxn[nt - 8] = wmma4(af, bi, gxn[nt - 8]);
                    ghn[nt - 8] = wmma4(ah, bh, ghn[nt - 8]);
                }
            }
        }

        // element-wise gates + h update; write h back to LDS in A-source form
#pragma unroll
        for (int nt = 0; nt < 4; ++nt) {
#pragma unroll
            for (int e = 0; e < 8; ++e) {
                const float rv = sigmoidf_(arz[nt][e]     + brz[nt]);
                const float zv = sigmoidf_(arz[4 + nt][e] + brz[4 + nt]);
                const float nv = tanhf(gxn[nt][e] + bin_[nt] +
                                       rv * (ghn[nt][e] + bhn_[nt]));
                const float hv = (1.0f - zv) * nv + zv * h[nt][e];
                h[nt][e] = hv;
                hld[(e + 8 * half) * WPAD + nt * 16 + l15] = hv;
            }
        }
    }

    // src = h @ Wo^T + bo
    v8f os[4];
#pragma unroll
    for (int nt = 0; nt < 4; ++nt) os[nt] = v8zero();
    for (int kb = 0; kb < DDIM; kb += 4) {
        const int k = kb + koff;
        v2f a = *(const v2f*)(hld + l15 * WPAD + k);
#pragma unroll
        for (int nt = 0; nt < 4; ++nt) {
            v2f b = *(const v2f*)(Wo + (nt * 16 + l15) * DDIM + k);
            os[nt] = wmma4(a, b, os[nt]);
        }
    }
#pragma unroll
    for (int nt = 0; nt < 4; ++nt) {
        const int d  = nt * 16 + l15;
        const float bb = bo[d];
#pragma unroll
        for (int e = 0; e < 8; ++e) {
            const int m = e + 8 * half;
            out_src[(long long)(base + m) * DDIM + d] = os[nt][e] + bb;
        }
    }
}

// =====================================================================
// Kernel C: dst = (node_raw[dst_ids] @ Wf^T + bf) @ Wo^T + bo
// One wave per 16-batch tile; blockDim = 128 (4 waves).
// =====================================================================
__global__ void __launch_bounds__(128)
dst_proj_kernel(const float* __restrict__ node_raw,
                const int*   __restrict__ dst_ids,
                const float* __restrict__ Wf, const float* __restrict__ bf,
                const float* __restrict__ Wo, const float* __restrict__ bo,
                float* __restrict__ out_dst)
{
    __shared__ float eld_all[4 * 16 * WPAD];
    const int tid  = threadIdx.x;
    const int lane = tid & 31;
    const int wv   = tid >> 5;
    const int l15  = lane & 15;
    const int half = lane >> 4;
    const int koff = 2 * half;
    float* eld = eld_all + wv * (16 * WPAD);

    const int base = (blockIdx.x * 4 + wv) * 16;
    const long long did = dst_ids[base + l15];
    const float* x = node_raw + did * NFEAT;

    v8f a1[4];
#pragma unroll
    for (int nt = 0; nt < 4; ++nt) a1[nt] = v8zero();
    for (int kb = 0; kb < NFEAT; kb += 4) {
        v2f a = *(const v2f*)(x + kb + koff);
#pragma unroll
        for (int nt = 0; nt < 4; ++nt) {
            v2f b = *(const v2f*)(Wf + (nt * 16 + l15) * NFEAT + kb + koff);
            a1[nt] = wmma4(a, b, a1[nt]);
        }
    }
    // stage embedding (+ bf) in LDS as A-operand source
#pragma unroll
    for (int nt = 0; nt < 4; ++nt) {
        const float bb = bf[nt * 16 + l15];
#pragma unroll
        for (int e = 0; e < 8; ++e)
            eld[(e + 8 * half) * WPAD + nt * 16 + l15] = a1[nt][e] + bb;
    }
    __syncthreads();

    v8f a2[4];
#pragma unroll
    for (int nt = 0; nt < 4; ++nt) a2[nt] = v8zero();
    for (int kb = 0; kb < DDIM; kb += 4) {
        const int k = kb + koff;
        v2f a = *(const v2f*)(eld + l15 * WPAD + k);
#pragma unroll
        for (int nt = 0; nt < 4; ++nt) {
            v2f b = *(const v2f*)(Wo + (nt * 16 + l15) * DDIM + k);
            a2[nt] = wmma4(a, b, a2[nt]);
        }
    }
#pragma unroll
    for (int nt = 0; nt < 4; ++nt) {
        const int d  = nt * 16 + l15;
        const float bb = bo[d];
#pragma unroll
        for (int e = 0; e < 8; ++e) {
            const int m = e + 8 * half;
            out_dst[(long long)(base + m) * DDIM + d] = a2[nt][e] + bb;
        }
    }
}

// =====================================================================
extern "C" void kernel_launch(void* const* d_in, const int* in_sizes, int n_in,
                              void* d_out, int out_size, void* d_ws, size_t ws_size,
                              hipStream_t stream) {
    const float* node_raw = (const float*)d_in[0];
    const float* edge_raw = (const float*)d_in[1];
    const int*   nids     = (const int*)  d_in[2];
    const int*   eids     = (const int*)  d_in[3];
    const float* times    = (const float*)d_in[4];
    const int*   dst_ids  = (const int*)  d_in[5];
    const float* Wf  = (const float*)d_in[6];
    const float* bf  = (const float*)d_in[7];
    const float* We  = (const float*)d_in[8];
    const float* be  = (const float*)d_in[9];
    const float* Wt  = (const float*)d_in[10];
    const float* bt  = (const float*)d_in[11];
    const float* Ws  = (const float*)d_in[12];
    const float* bs  = (const float*)d_in[13];
    const float* Wo  = (const float*)d_in[14];
    const float* bo  = (const float*)d_in[15];
    const float* Wih = (const float*)d_in[16];
    const float* bih = (const float*)d_in[17];
    const float* Whh = (const float*)d_in[18];
    const float* bhh = (const float*)d_in[19];
    const float* tw  = (const float*)d_in[20];
    const float* tb  = (const float*)d_in[21];

    float* fused   = (float*)d_ws;                  // [B*NB, 64] = 52.4 MB
    float* out_src = (float*)d_out;                 // [B, 64]
    float* out_dst = out_src + (size_t)BSZ * DDIM;  // [B, 64]

    // Kernel A: 204800/16 = 12800 wave-tiles, 4 waves/block
    embed_fuse_kernel<<<M1 / 16 / 4, 128, 0, stream>>>(
        node_raw, edge_raw, nids, eids, times, dst_ids,
        Wf, bf, We, be, Wt, bt, Ws, bs, tw, tb, fused);

    // Kernel B: 4096/(4*16) = 64 blocks; dynamic LDS for weights + h tiles
    const size_t shB = (size_t)(WLDS_FLOATS + 4 * 16 * WPAD) * sizeof(float);
    gru_out_kernel<<<BSZ / 64, 128, shB, stream>>>(
        fused, Wih, bih, Whh, bhh, Wo, bo, out_src);

    // Kernel C: 4096/(4*16) = 64 blocks
    dst_proj_kernel<<<BSZ / 64, 128, 0, stream>>>(
        node_raw, dst_ids, Wf, bf, Wo, bo, out_dst);
}